// MyLiNet_58179626992175
// MI455X (gfx1250) — compile-verified
//
#include <hip/hip_runtime.h>
#include <hip/hip_bf16.h>
#include <stddef.h>

// Problem constants (from reference)
#define N0 65536
#define N1 16384
#define K0 32
#define N2 150
#define N2P 152   // N2 padded to a multiple of 4 (WMMA K-step)
#define K1 64
#define BB 64     // batch

typedef __attribute__((ext_vector_type(2))) float v2f;
typedef __attribute__((ext_vector_type(8))) float v8f;

// ---------------------------------------------------------------------------
// Kernel 1: transpose x (B x N0, row-major) -> xT (N0 x B, row-major)
// so a gathered "column" x[:, idx] becomes a contiguous 256B row xT[idx, :].
// LDS-tiled transpose; grid (N0/32, B/32), block (32, 8).
// ---------------------------------------------------------------------------
__global__ void transpose_x_kernel(const float* __restrict__ x,
                                   float* __restrict__ xT) {
    __shared__ float tile[32][33];
    const int jt = blockIdx.x * 32;   // N0 tile base
    const int bt = blockIdx.y * 32;   // batch tile base
    const int tx = threadIdx.x;       // 0..31
    const int ty = threadIdx.y;       // 0..7
#pragma unroll
    for (int r = 0; r < 32; r += 8) {
        tile[ty + r][tx] = x[(size_t)(bt + ty + r) * N0 + (jt + tx)];
    }
    __syncthreads();
#pragma unroll
    for (int r = 0; r < 32; r += 8) {
        xT[(size_t)(jt + ty + r) * BB + (bt + tx)] = tile[tx][ty + r];
    }
}

// ---------------------------------------------------------------------------
// Kernel 2: sparse layer 0. One wave32 per neuron; each lane owns 2 batch
// elements (float2 over the contiguous 64-float xT row). conn/W row addresses
// are wave-uniform (readfirstlane) -> scalar SMEM loads.
// hT layout: (N1 x B) row-major so layer 1's gathers are contiguous too.
// ---------------------------------------------------------------------------
__global__ void layer0_kernel(const float* __restrict__ xT,
                              const int*   __restrict__ conn0,
                              const float* __restrict__ W0,
                              const float* __restrict__ b0,
                              float*       __restrict__ hT) {
    const int lane = threadIdx.x & 31;
    const int wave = threadIdx.x >> 5;          // 8 waves / block
    int n = blockIdx.x * 8 + wave;              // 0..N1-1 (grid = N1/8)
    n = __builtin_amdgcn_readfirstlane(n);      // wave-uniform -> s_load rows

    const int*   ci = conn0 + (size_t)n * K0;
    const float* wi = W0    + (size_t)n * K0;

    float accx = 0.0f, accy = 0.0f;
#pragma unroll
    for (int k = 0; k < K0; ++k) {
        const int   idx = ci[k];                // uniform: scalar load
        const float w   = wi[k];                // uniform: scalar load
        const v2f v = *(const v2f*)(xT + (size_t)idx * BB + lane * 2);
        accx = fmaf(w, v.x, accx);
        accy = fmaf(w, v.y, accy);
    }
    const float bias = b0[n];
    v2f r;
    r.x = fmaxf(accx + bias, 0.0f);
    r.y = fmaxf(accy + bias, 0.0f);
    *(v2f*)(hT + (size_t)n * BB + lane * 2) = r;
}

// ---------------------------------------------------------------------------
// Kernel 3: sparse layer 1 (N2=150 neurons, K1=64). Same scheme; output h2
// stored (B x N2P) row-major.  Grid is 19 blocks x 8 waves = 152 wave slots:
// slots 150/151 write the zero pad columns so the FC kernel needs no guards.
// ---------------------------------------------------------------------------
__global__ void layer1_kernel(const float* __restrict__ hT,
                              const int*   __restrict__ conn1,
                              const float* __restrict__ W1,
                              const float* __restrict__ b1,
                              float*       __restrict__ h2) {
    const int lane = threadIdx.x & 31;
    const int wave = threadIdx.x >> 5;
    int n = blockIdx.x * 8 + wave;              // 0..151
    n = __builtin_amdgcn_readfirstlane(n);

    if (n < N2) {
        const int*   ci = conn1 + (size_t)n * K1;
        const float* wi = W1    + (size_t)n * K1;

        float accx = 0.0f, accy = 0.0f;
#pragma unroll 8
        for (int k = 0; k < K1; ++k) {
            const int   idx = ci[k];
            const float w   = wi[k];
            const v2f v = *(const v2f*)(hT + (size_t)idx * BB + lane * 2);
            accx = fmaf(w, v.x, accx);
            accy = fmaf(w, v.y, accy);
        }
        const float bias = b1[n];
        h2[(size_t)(2 * lane)     * N2P + n] = fmaxf(accx + bias, 0.0f);
        h2[(size_t)(2 * lane + 1) * N2P + n] = fmaxf(accy + bias, 0.0f);
    } else {
        // zero pad columns 150..151 for every batch row
        h2[(size_t)(2 * lane)     * N2P + n] = 0.0f;
        h2[(size_t)(2 * lane + 1) * N2P + n] = 0.0f;
    }
}

// ---------------------------------------------------------------------------
// Kernel 4: FC head with WMMA.  out(64x2) = h2(64x150) @ fc_w^T(150x2) + fc_b
// V_WMMA_F32_16X16X4_F32: D(16x16) = A(16x4) x B(4x16) + C, f32 throughout.
// 1 block, 4 waves; wave w handles batch rows [16w, 16w+16).  K padded to
// 152 (zeros already materialized in h2); N padded 2 -> 16 via an LDS-staged
// zero-padded B matrix, so the inner loop is branch-free:
//     global_load_b64 (A frag) + ds_load_b64 (B frag) + v_wmma.
//
// Per-lane fragment layout (ISA 7.12.2):
//   A 16x4 : m = lane&15, half = lane>>4; v = { A[m][2*half], A[m][2*half+1] }
//   B 4x16 : n = lane&15;                 v = { B[2*half][n], B[2*half+1][n] }
//   C/D    : n = lane&15; vgpr i = D[i + 8*half][n]
// ---------------------------------------------------------------------------
__global__ void fc_wmma_kernel(const float* __restrict__ h2,
                               const float* __restrict__ fc_w,
                               const float* __restrict__ fc_b,
                               float*       __restrict__ out) {
    // Bp[n][k] = fc_w[n][k] zero-padded to 16 x 152 (== fc_w^T as B-matrix)
    __shared__ float Bp[16 * N2P];
    for (int i = threadIdx.x; i < 16 * N2P; i += 128) {
        const int n = i / N2P;
        const int k = i - n * N2P;
        Bp[i] = (n < 2 && k < N2) ? fc_w[n * N2 + k] : 0.0f;
    }
    __syncthreads();

    const int lane = threadIdx.x & 31;
    const int wave = threadIdx.x >> 5;     // 0..3
    const int b0   = wave * 16;            // batch tile base
    const int m    = lane & 15;
    const int half = lane >> 4;            // 0 or 1

    const float* arow = h2 + (size_t)(b0 + m) * N2P + half * 2;
    const float* brow = Bp + (size_t)m * N2P + half * 2;

    v8f c = {};
#pragma unroll 2
    for (int kk = 0; kk < N2P / 4; ++kk) {
        const v2f a = *(const v2f*)(arow + kk * 4);   // 8B-aligned global
        const v2f b = *(const v2f*)(brow + kk * 4);   // 8B-aligned LDS
        // 8 args: (neg_a, A, neg_b, B, c_mod, C, reuse_a, reuse_b)
        c = __builtin_amdgcn_wmma_f32_16x16x4_f32(
                false, a, false, b, (short)0, c, false, false);
    }

    // Lane holds D[i + 8*half][m] for i = 0..7; only columns m<2 are real.
    if (m < 2) {
        const float bias = fc_b[m];
#pragma unroll
        for (int i = 0; i < 8; ++i) {
            const int brow_idx = b0 + i + 8 * half;
            out[brow_idx * 2 + m] = c[i] + bias;
        }
    }
}

// ---------------------------------------------------------------------------
// Launch: inputs in setup_inputs() order:
//   0:x 1:conn0 2:conn1 3:W0 4:b0 5:W1 6:b1 7:fc_w 8:fc_b
// Workspace: xT (16.78MB) | hT (4.19MB) | h2 (38.9KB)  (~21MB total)
// ---------------------------------------------------------------------------
extern "C" void kernel_launch(void* const* d_in, const int* in_sizes, int n_in,
                              void* d_out, int out_size, void* d_ws, size_t ws_size,
                              hipStream_t stream) {
    const float* x     = (const float*)d_in[0];
    const int*   conn0 = (const int*)  d_in[1];
    const int*   conn1 = (const int*)  d_in[2];
    const float* W0    = (const float*)d_in[3];
    const float* b0    = (const float*)d_in[4];
    const float* W1    = (const float*)d_in[5];
    const float* b1    = (const float*)d_in[6];
    const float* fc_w  = (const float*)d_in[7];
    const float* fc_b  = (const float*)d_in[8];
    float*       out   = (float*)d_out;

    char* ws = (char*)d_ws;
    float* xT = (float*)ws;                                           // N0*B
    float* hT = (float*)(ws + (size_t)N0 * BB * sizeof(float));       // N1*B
    float* h2 = (float*)(ws + (size_t)N0 * BB * sizeof(float)
                            + (size_t)N1 * BB * sizeof(float));       // B*N2P

    transpose_x_kernel<<<dim3(N0 / 32, BB / 32), dim3(32, 8), 0, stream>>>(x, xT);
    layer0_kernel<<<dim3(N1 / 8), dim3(256), 0, stream>>>(xT, conn0, W0, b0, hT);
    layer1_kernel<<<dim3(N2P / 8), dim3(256), 0, stream>>>(hT, conn1, W1, b1, h2);
    fc_wmma_kernel<<<dim3(1), dim3(128), 0, stream>>>(h2, fc_w, fc_b, out);
}